// CosineSimilarityLoss_54434415509810
// MI455X (gfx1250) — compile-verified
//
#include <hip/hip_runtime.h>

#define PEAKS 256
#define NBINS 2000
#define FEPS 1e-8f

// ---- CDNA5 async global->LDS path (probe with __has_builtin) --------------
#if defined(__has_builtin)
# if __has_builtin(__builtin_amdgcn_global_load_async_to_lds_b32)
#  define HAVE_ASYNC_LDS 1
# endif
# if __has_builtin(__builtin_amdgcn_s_wait_asynccnt)
#  define HAVE_WAIT_ASYNC 1
# endif
#endif

typedef __attribute__((address_space(3))) int lint_t;

__device__ __forceinline__ void g2l_async_b32(const float* g, float* l) {
#if defined(HAVE_ASYNC_LDS)
  __builtin_amdgcn_global_load_async_to_lds_b32((int*)g, (lint_t*)l, 0, 0);
#else
  *l = *g;  // synchronous fallback
#endif
}

__device__ __forceinline__ void wait_async0() {
#if defined(HAVE_ASYNC_LDS)
# if defined(HAVE_WAIT_ASYNC)
  __builtin_amdgcn_s_wait_asynccnt(0);
# else
  asm volatile("s_wait_asynccnt 0" ::: "memory");
# endif
#endif
}

// ---- Kernel 1: one workgroup (256 thr = 8 waves) per batch row -------------
// LDS: 2 x 2000-bin histograms (16KB) + 6 x 256 staged inputs (6KB) + 3KB red.
extern "C" __global__ void __launch_bounds__(PEAKS)
spec_cos_rows(const float* __restrict__ pmz,  const float* __restrict__ pint,
              const float* __restrict__ pconf, const float* __restrict__ tmz,
              const float* __restrict__ tint, const float* __restrict__ tmask,
              float* __restrict__ row_cos)
{
  __shared__ float sP[NBINS];
  __shared__ float sT[NBINS];
  __shared__ float stage[6 * PEAKS];
  __shared__ float red0[PEAKS], red1[PEAKS], red2[PEAKS];

  const int t = threadIdx.x;
  const long long base = (long long)blockIdx.x * PEAKS + t;

  // Kick off async staging of all 6 input elements for this thread...
  g2l_async_b32(pmz   + base, &stage[0 * PEAKS + t]);
  g2l_async_b32(pint  + base, &stage[1 * PEAKS + t]);
  g2l_async_b32(pconf + base, &stage[2 * PEAKS + t]);
  g2l_async_b32(tmz   + base, &stage[3 * PEAKS + t]);
  g2l_async_b32(tint  + base, &stage[4 * PEAKS + t]);
  g2l_async_b32(tmask + base, &stage[5 * PEAKS + t]);

  // ...and overlap with zeroing both histograms.
  for (int i = t; i < NBINS; i += PEAKS) { sP[i] = 0.0f; sT[i] = 0.0f; }

  wait_async0();
  __syncthreads();

  // Scatter: weighted-intensity histogram via LDS float atomics (ds_add_f32).
  {
    float mzp = stage[0 * PEAKS + t];
    float wp  = stage[1 * PEAKS + t] * stage[2 * PEAKS + t];
    int bp = (int)(mzp * 2000.0f);
    bp = bp < 0 ? 0 : (bp > NBINS - 1 ? NBINS - 1 : bp);
    atomicAdd(&sP[bp], wp);

    float mzt = stage[3 * PEAKS + t];
    float wt  = stage[4 * PEAKS + t] * stage[5 * PEAKS + t];
    int bt = (int)(mzt * 2000.0f);
    bt = bt < 0 ? 0 : (bt > NBINS - 1 ? NBINS - 1 : bt);
    atomicAdd(&sT[bt], wt);
  }
  __syncthreads();

  // Fused reduction: dot, ||p||^2, ||t||^2 over the raw histograms.
  float dot = 0.0f, pp = 0.0f, tt = 0.0f;
  for (int i = t; i < NBINS; i += PEAKS) {
    float a = sP[i], b = sT[i];
    dot = fmaf(a, b, dot);
    pp  = fmaf(a, a, pp);
    tt  = fmaf(b, b, tt);
  }
  red0[t] = dot; red1[t] = pp; red2[t] = tt;
  __syncthreads();
  for (int s = PEAKS / 2; s > 0; s >>= 1) {
    if (t < s) {
      red0[t] += red0[t + s];
      red1[t] += red1[t + s];
      red2[t] += red2[t + s];
    }
    __syncthreads();
  }

  if (t == 0) {
    // Reference math: binned/(norm+eps) normalization, then cosine with
    // max(n, eps) guards.  na = np/(np+eps), nb = nt/(nt+eps).
    float np = sqrtf(red1[0]);
    float nt = sqrtf(red2[0]);
    float ndot = red0[0] / ((np + FEPS) * (nt + FEPS));
    float na = np / (np + FEPS);
    float nb = nt / (nt + FEPS);
    float c = ndot / (fmaxf(na, FEPS) * fmaxf(nb, FEPS));
    row_cos[blockIdx.x] = c;
  }
}

// ---- Kernel 2: deterministic tree reduction of per-row cos -> scalar -------
extern "C" __global__ void __launch_bounds__(256)
spec_cos_finalize(const float* __restrict__ row_cos, float* __restrict__ out, int B)
{
  __shared__ float red[256];
  int t = threadIdx.x;
  float s = 0.0f;
  for (int i = t; i < B; i += 256) s += row_cos[i];
  red[t] = s;
  __syncthreads();
  for (int k = 128; k > 0; k >>= 1) {
    if (t < k) red[t] += red[t + k];
    __syncthreads();
  }
  if (t == 0) out[0] = 1.0f - red[0] / (float)B;  // WEIGHT = 1.0
}

// ---- Host entry ------------------------------------------------------------
extern "C" void kernel_launch(void* const* d_in, const int* in_sizes, int n_in,
                              void* d_out, int out_size, void* d_ws, size_t ws_size,
                              hipStream_t stream)
{
  const float* pmz   = (const float*)d_in[0];
  const float* pint  = (const float*)d_in[1];
  const float* pconf = (const float*)d_in[2];
  const float* tmz   = (const float*)d_in[3];
  const float* tint  = (const float*)d_in[4];
  const float* tmask = (const float*)d_in[5];

  const int B = in_sizes[0] / PEAKS;      // 16384
  float* row_cos = (float*)d_ws;          // B floats of scratch

  spec_cos_rows<<<B, PEAKS, 0, stream>>>(pmz, pint, pconf, tmz, tint, tmask, row_cos);
  spec_cos_finalize<<<1, 256, 0, stream>>>(row_cos, (float*)d_out, B);
}